// RepresentationNgramEstimator_45105746542759
// MI455X (gfx1250) — compile-verified
//
#include <hip/hip_runtime.h>

// Problem constants (from reference): B=32, L=4096, D=128, N=5, V=128
#define Bsz  32
#define Lseq 4096
#define Dd   128
#define Vv   128
#define NCTX 4            // N-1 context slots
#define Wf   512          // D*(N-1) = W row stride

typedef __attribute__((ext_vector_type(2))) float v2f;
typedef __attribute__((ext_vector_type(8))) float v8f;

// ---------------------------------------------------------------------------
// Kernel 1: precompute M[k][t][v] = sum_d emb[t,d] * W[v, k*128+d]
// via fp32 WMMA (V_WMMA_F32_16X16X4_F32). 4 slots x (8x8) 16x16 tiles = 256
// waves, each wave does 32 unrolled K=4 WMMAs. EXEC all-ones (blockDim=32).
//
// fp32 WMMA VGPR layouts (ISA 7.12.2):
//   A 16x4 : lane(0..15)->M, halves hold K={0,1} / {2,3} in VGPR0/1
//   B 4x16 : lane(0..15)->N, halves hold K={0,1} / {2,3} (mirror of A)
//   C 16x16: VGPR r -> M = r + 8*half, N = lane & 15
// ---------------------------------------------------------------------------
__global__ __launch_bounds__(32) void ngram_table_wmma(
    const float* __restrict__ emb,   // (V, D)   = (128,128)
    const float* __restrict__ W,     // (V, 512)
    float* __restrict__ Mtab)        // (4, 128, 128)
{
    const int wid  = blockIdx.x;        // 0..255
    const int k    = wid >> 6;          // context slot 0..3
    const int tile = wid & 63;
    const int m0   = (tile >> 3) << 4;  // token-row tile base
    const int n0   = (tile & 7)  << 4;  // v-col tile base

    const int lane = threadIdx.x;       // 0..31
    const int half = lane >> 4;
    const int lr   = lane & 15;

    const float* arow = emb + (m0 + lr) * Dd;         // A: row M = lr
    const float* brow = W   + (n0 + lr) * Wf + k * Dd; // B: col N = lr (W is row-major over v)

    v8f c = {};
#pragma unroll
    for (int kk = 0; kk < 32; ++kk) {
        const int d0 = kk * 4 + half * 2;
        v2f a, b;
        a.x = arow[d0 + 0];   // K = 2*half
        a.y = arow[d0 + 1];   // K = 2*half + 1
        b.x = brow[d0 + 0];
        b.y = brow[d0 + 1];
        // 8-arg form: (neg_a, A, neg_b, B, c_mod, C, reuse_a, reuse_b)
        c = __builtin_amdgcn_wmma_f32_16x16x4_f32(
                false, a, false, b, (short)0, c, false, false);
    }

    float* outTile = Mtab + k * (Vv * Vv);
#pragma unroll
    for (int r = 0; r < 8; ++r) {
        const int row = m0 + r + 8 * half;   // token index
        outTile[row * Vv + (n0 + lr)] = c[r];
    }
}

// ---------------------------------------------------------------------------
// Kernel 2 (bandwidth bound): one wave per (b,j) position.
// out[b,j,v] = bias[v] + sum_k M[k][x[b,j-4+k]][v]   (bias only for j<4)
// Each lane covers 4 consecutive v via float4 (global_load_b128 /
// global_store_b128). M tables (256 KB) stay resident in the 192 MB L2, so
// HBM traffic ~= the 64 MiB output write -> ~2.8 us floor at 23.3 TB/s.
// ---------------------------------------------------------------------------
__global__ __launch_bounds__(256) void ngram_gather_sum(
    const int*   __restrict__ x,      // (B, L)
    const float* __restrict__ bias,   // (V,)
    const float* __restrict__ Mtab,   // (4, 128, 128)
    float4*      __restrict__ out)    // (B, L, V) as float4
{
    const int lane = threadIdx.x & 31;
    const int wave = threadIdx.x >> 5;
    const int pos  = blockIdx.x * 8 + wave;   // 0 .. B*L-1
    const int b    = pos >> 12;               // / 4096
    const int j    = pos & (Lseq - 1);

    float4 acc = ((const float4*)bias)[lane];

    if (j >= NCTX) {
        const int* xp = x + b * Lseq + (j - NCTX);
#pragma unroll
        for (int k = 0; k < NCTX; ++k) {
            const int t = xp[k];  // uniform across wave -> broadcast-friendly
            const float4 m =
                ((const float4*)(Mtab + (k * (Vv * Vv) + t * Vv)))[lane];
            acc.x += m.x; acc.y += m.y; acc.z += m.z; acc.w += m.w;
        }
    }
    out[pos * (Vv / 4) + lane] = acc;
}

// ---------------------------------------------------------------------------
extern "C" void kernel_launch(void* const* d_in, const int* in_sizes, int n_in,
                              void* d_out, int out_size, void* d_ws, size_t ws_size,
                              hipStream_t stream) {
    const int*   x    = (const int*)  d_in[0];  // (B,L) int32
    const float* emb  = (const float*)d_in[1];  // (V,D)
    const float* W    = (const float*)d_in[2];  // (V,512)
    const float* bias = (const float*)d_in[3];  // (V,)

    float* Mtab = (float*)d_ws;  // needs 4*128*128*4 = 256 KiB scratch

    // Stage 1: 256 waves of fp32 WMMA build the 4 token->logit tables.
    ngram_table_wmma<<<256, 32, 0, stream>>>(emb, W, Mtab);

    // Stage 2: one wave per sequence position, float4-vectorized.
    const int positions = Bsz * Lseq;           // 131072
    ngram_gather_sum<<<positions / 8, 256, 0, stream>>>(
        x, bias, Mtab, (float4*)d_out);
}